// Model_52381421142377
// MI455X (gfx1250) — compile-verified
//
#include <hip/hip_runtime.h>
#include <hip/hip_bf16.h>

// ---------------------------------------------------------------------------
// Problem constants (from reference)
// ---------------------------------------------------------------------------
#define B_SAMP 4096
#define N_NODES 90
#define T_LEN 195
#define HID 20
#define NP 96    // N padded to multiple of 16
#define TP 196   // T padded to multiple of 4 (WMMA K step)
#define HP 32    // HID padded to multiple of 16
#define EPS 1e-5f

typedef __attribute__((ext_vector_type(2))) float v2f;
typedef __attribute__((ext_vector_type(8))) float v8f;
typedef __attribute__((ext_vector_type(4))) unsigned int v4u;
typedef __attribute__((ext_vector_type(4))) int v4i;
typedef __attribute__((ext_vector_type(8))) int v8i;

// ---------------------------------------------------------------------------
// WMMA helpers: V_WMMA_F32_16X16X4_F32, wave32 layouts per CDNA5 ISA
//   A (16x4, 32-bit): lanes 0-15 hold M=lane, K={0,1}; lanes 16-31 K={2,3}
//   B (4x16):         lanes 0-15 hold N=lane, K={0,1}; lanes 16-31 K={2,3}
//   C/D (16x16):      VGPR r: lanes 0-15 -> (M=r, N=lane); lanes 16-31 -> M=r+8
// ---------------------------------------------------------------------------
__device__ __forceinline__ v8f wmma4(v2f a, v2f b, v8f c) {
  return __builtin_amdgcn_wmma_f32_16x16x4_f32(false, a, false, b, (short)0, c,
                                               false, false);
}

__device__ __forceinline__ v2f ld_a(const float* s, int ld, int lane) {
  int m = lane & 15, kb = (lane >> 4) << 1;
  v2f a;
  a.x = s[m * ld + kb];
  a.y = s[m * ld + kb + 1];
  return a;
}

__device__ __forceinline__ v2f ld_b(const float* s, int ld, int lane) {
  int n = lane & 15, kb = (lane >> 4) << 1;
  v2f b;
  b.x = s[kb * ld + n];
  b.y = s[(kb + 1) * ld + n];
  return b;
}

// B tile taken from a row-major matrix used *transposed* (for X @ X^T)
__device__ __forceinline__ v2f ld_bT(const float* s, int ld, int lane) {
  int n = lane & 15, kb = (lane >> 4) << 1;
  v2f b;
  b.x = s[n * ld + kb];
  b.y = s[n * ld + kb + 1];
  return b;
}

// B tile loaded straight from a (small, cache-resident) global matrix with
// bounds guards providing the zero padding.
__device__ __forceinline__ v2f ld_b_g(const float* __restrict__ g, int ldg,
                                      int rows, int cols, int k0, int c0,
                                      int lane) {
  int n = lane & 15, kb = (lane >> 4) << 1;
  int r = k0 + kb, c = c0 + n;
  v2f b;
  b.x = (r < rows && c < cols) ? g[r * ldg + c] : 0.f;
  b.y = (r + 1 < rows && c < cols) ? g[(r + 1) * ldg + c] : 0.f;
  return b;
}

__device__ __forceinline__ void st_c(float* d, int ld, v8f c, int lane) {
  int n = lane & 15, mb = (lane >> 4) << 3;
#pragma unroll
  for (int r = 0; r < 8; ++r) d[(mb + r) * ld + n] = c[r];
}

// ---------------------------------------------------------------------------
// Tensor Data Mover: 2-D tile load Global -> LDS (packed row-major in LDS).
// tensor_dim0/1 are the *remaining* extent from the tile origin so OOB rows /
// cols of the tile are zero-filled by hardware (free K-tail handling).
// ---------------------------------------------------------------------------
__device__ __forceinline__ void tdm_load_2d(unsigned int lds_off,
                                            const void* gaddr,
                                            unsigned int tensor_d0,
                                            unsigned int tensor_d1,
                                            unsigned int tile_d0,
                                            unsigned int tile_d1,
                                            unsigned int stride_elems) {
  unsigned long long ga = (unsigned long long)(uintptr_t)gaddr;
  v4u g0;
  g0.x = 1u;                    // count=1, user descriptor
  g0.y = lds_off;               // lds_addr (bytes)
  g0.z = (unsigned int)ga;      // global_addr[31:0]
  g0.w = (unsigned int)((ga >> 32) & 0x1FFFFFFull) | (2u << 30);  // [56:32]|type=2
  v8i g1;
  g1[0] = (int)(2u << 16);                                   // data_size = 4B
  g1[1] = (int)((tensor_d0 & 0xFFFFu) << 16);                // tensor_dim0 lo
  g1[2] = (int)((tensor_d0 >> 16) | ((tensor_d1 & 0xFFFFu) << 16));
  g1[3] = (int)((tensor_d1 >> 16) | (tile_d0 << 16));        // tile_dim0
  g1[4] = (int)(tile_d1 & 0xFFFFu);                          // tile_dim1, dim2=0
  g1[5] = (int)stride_elems;                                 // dim0_stride lo32
  g1[6] = 0;
  g1[7] = 0;
  v4i z4 = {0, 0, 0, 0};
#if __clang_major__ >= 23
  v8i z8 = {0, 0, 0, 0, 0, 0, 0, 0};
  __builtin_amdgcn_tensor_load_to_lds(g0, g1, z4, z4, z8, 0);
#else
  __builtin_amdgcn_tensor_load_to_lds(g0, g1, z4, z4, 0);
#endif
}

// ---------------------------------------------------------------------------
// Kernel 1: per-sample correlation + 2 graph convolutions, fully in LDS.
// 256 threads (8 wave32) per sample; ~137 KB LDS -> 2 workgroups per WGP.
// adj never touches HBM (saves ~400 MB of adjacency traffic).
// ---------------------------------------------------------------------------
__global__ __launch_bounds__(256) void gcn_corr_kernel(
    const float* __restrict__ input, const float* __restrict__ gc1,
    const float* __restrict__ gc2, float* __restrict__ flat) {
  __shared__ float xs[NP * TP];    // 73.5 KB  zero-padded sample [96][196]
  __shared__ float adjs[NP * NP];  // 36.0 KB  Gram -> correlation [96][96]
  __shared__ float xgs[NP * HP];   // 12.0 KB
  __shared__ float hs[NP * HP];    // 12.0 KB
  __shared__ float rs_[NP];        // row sums
  __shared__ float di_[NP];        // 1/sqrt(diag cov)

  const int tid = threadIdx.x;
  const int lane = tid & 31;
  const int wave = tid >> 5;
  const int b = blockIdx.x;
  const float* xin = input + (size_t)b * (N_NODES * T_LEN);

  // ---- cooperative zero-padded load of the sample ----
  for (int i = tid; i < NP * TP; i += 256) {
    int r = i / TP, c = i - r * TP;
    xs[i] = (r < N_NODES && c < T_LEN) ? xin[r * T_LEN + c] : 0.f;
  }
  __syncthreads();

  // ---- row sums (mean subtraction via cov = G - rs*rs^T/T) ----
  if (tid < NP) {
    float s = 0.f;
    for (int c = 0; c < T_LEN; ++c) s += xs[tid * TP + c];
    rs_[tid] = s;
  }
  __syncthreads();

  // ---- Gram G = X @ X^T  (96x96, 49 k-steps, 36 tiles over 8 waves) ----
  for (int t = wave; t < 36; t += 8) {
    int tr = t / 6, tc = t - tr * 6;
    v8f acc = {0.f, 0.f, 0.f, 0.f, 0.f, 0.f, 0.f, 0.f};
    const float* Ap = xs + tr * 16 * TP;
    const float* Bp = xs + tc * 16 * TP;
    for (int k = 0; k < TP / 4; ++k) {
      acc = wmma4(ld_a(Ap + k * 4, TP, lane), ld_bT(Bp + k * 4, TP, lane), acc);
    }
    st_c(adjs + tr * 16 * NP + tc * 16, NP, acc, lane);
  }
  __syncthreads();

  // ---- diag -> rsqrt; padded rows get 0 (kills padded adj entries) ----
  if (tid < NP) {
    float cnn = adjs[tid * NP + tid] - rs_[tid] * rs_[tid] * (1.f / T_LEN);
    di_[tid] = (tid < N_NODES) ? rsqrtf(cnn) : 0.f;
  }
  __syncthreads();

  // ---- adj = clip(cov / (d d^T), -1, 1) in place ----
  for (int i = tid; i < NP * NP; i += 256) {
    int n = i / NP, m = i - n * NP;
    float cov = adjs[i] - rs_[n] * rs_[m] * (1.f / T_LEN);
    float v = cov * di_[n] * di_[m];
    adjs[i] = fminf(fmaxf(v, -1.f), 1.f);
  }
  __syncthreads();

  // ---- xg = X @ W1  (96x32, K=196; W1 read from global, L2-resident) ----
  for (int t = wave; t < 12; t += 8) {
    int tr = t >> 1, tc = t & 1;
    v8f acc = {0.f, 0.f, 0.f, 0.f, 0.f, 0.f, 0.f, 0.f};
    for (int k = 0; k < TP / 4; ++k) {
      acc = wmma4(ld_a(xs + tr * 16 * TP + k * 4, TP, lane),
                  ld_b_g(gc1, HID, T_LEN, HID, k * 4, tc * 16, lane), acc);
    }
    st_c(xgs + tr * 16 * HP + tc * 16, HP, acc, lane);
  }
  __syncthreads();

  // ---- h = adj @ xg  (96x32, K=96) ----
  for (int t = wave; t < 12; t += 8) {
    int tr = t >> 1, tc = t & 1;
    v8f acc = {0.f, 0.f, 0.f, 0.f, 0.f, 0.f, 0.f, 0.f};
    for (int k = 0; k < NP / 4; ++k) {
      acc = wmma4(ld_a(adjs + tr * 16 * NP + k * 4, NP, lane),
                  ld_b(xgs + (k * 4) * HP + tc * 16, HP, lane), acc);
    }
    st_c(hs + tr * 16 * HP + tc * 16, HP, acc, lane);
  }
  __syncthreads();

  // ---- xg2 = h @ W2  (96x32, K=20 -> 5 k-steps; W2 from global) ----
  for (int t = wave; t < 12; t += 8) {
    int tr = t >> 1, tc = t & 1;
    v8f acc = {0.f, 0.f, 0.f, 0.f, 0.f, 0.f, 0.f, 0.f};
    for (int k = 0; k < 5; ++k) {
      acc = wmma4(ld_a(hs + tr * 16 * HP + k * 4, HP, lane),
                  ld_b_g(gc2, HID, HID, HID, k * 4, tc * 16, lane), acc);
    }
    st_c(xgs + tr * 16 * HP + tc * 16, HP, acc, lane);
  }
  __syncthreads();

  // ---- h2 = adj @ xg2  (96x32, K=96) ----
  for (int t = wave; t < 12; t += 8) {
    int tr = t >> 1, tc = t & 1;
    v8f acc = {0.f, 0.f, 0.f, 0.f, 0.f, 0.f, 0.f, 0.f};
    for (int k = 0; k < NP / 4; ++k) {
      acc = wmma4(ld_a(adjs + tr * 16 * NP + k * 4, NP, lane),
                  ld_b(xgs + (k * 4) * HP + tc * 16, HP, lane), acc);
    }
    st_c(hs + tr * 16 * HP + tc * 16, HP, acc, lane);
  }
  __syncthreads();

  // ---- write flat[b, n*20 + j] ----
  float* fo = flat + (size_t)b * (N_NODES * HID);
  for (int i = tid; i < N_NODES * HID; i += 256) {
    int n = i / HID, j = i - n * HID;
    fo[i] = hs[n * HP + j];
  }
}

// ---------------------------------------------------------------------------
// Kernel 2: tiled WMMA GEMM  C[M,N] = A[M,K] @ B[K,N] + bias
// 64x64 block tile, 256 threads = 8 waves x 2 WMMA tiles, K-tile = 16.
// A/B tiles are DMA'd by the Tensor Data Mover into double-buffered LDS,
// overlapped with WMMA on the other buffer (wave 0 drives the TDM; the
// readfirstlane trick makes the guard a *scalar* branch so EXEC stays all-1s
// and the EXEC-ignoring TDM op is issued exactly once).
// ---------------------------------------------------------------------------
#define BMT 64
#define BNT 64
#define KT 16

__global__ __launch_bounds__(256) void gemm_bias_wmma(
    const float* __restrict__ A, const float* __restrict__ Bm,
    float* __restrict__ C, int M, int N, int K,
    const float* __restrict__ bias) {
  __shared__ float As[2][BMT * KT];
  __shared__ float Bs[2][KT * BNT];

  const int tid = threadIdx.x;
  const int lane = tid & 31;
  const int wave0 = (__builtin_amdgcn_readfirstlane(tid) >> 5) == 0;
  const int wave = tid >> 5;
  const int bm = blockIdx.y * BMT;
  const int bn = blockIdx.x * BNT;

  const int t0 = wave * 2, t1 = wave * 2 + 1;
  const int tr0 = t0 >> 2, tc0 = t0 & 3;
  const int tr1 = t1 >> 2, tc1 = t1 & 3;

  v8f acc0 = {0.f, 0.f, 0.f, 0.f, 0.f, 0.f, 0.f, 0.f};
  v8f acc1 = {0.f, 0.f, 0.f, 0.f, 0.f, 0.f, 0.f, 0.f};

  const int nk = (K + KT - 1) / KT;

  if (wave0) {
    // prime buffer 0 with the k0 = 0 tiles
    tdm_load_2d((unsigned int)(uintptr_t)&As[0][0], A + (size_t)bm * K,
                (unsigned int)K, (unsigned int)(M - bm), KT, BMT,
                (unsigned int)K);
    tdm_load_2d((unsigned int)(uintptr_t)&Bs[0][0], Bm + bn,
                (unsigned int)(N - bn), (unsigned int)K, BNT, KT,
                (unsigned int)N);
  }

  for (int i = 0; i < nk; ++i) {
    const int cur = i & 1, nxt = cur ^ 1;
    if (wave0) __builtin_amdgcn_s_wait_tensorcnt(0);  // cur tiles landed
    __syncthreads();
    if (wave0 && (i + 1) < nk) {  // kick DMA for i+1 while we compute on i
      int k0 = (i + 1) * KT;
      tdm_load_2d((unsigned int)(uintptr_t)&As[nxt][0],
                  A + (size_t)bm * K + k0, (unsigned int)(K - k0),
                  (unsigned int)(M - bm), KT, BMT, (unsigned int)K);
      tdm_load_2d((unsigned int)(uintptr_t)&Bs[nxt][0],
                  Bm + (size_t)k0 * N + bn, (unsigned int)(N - bn),
                  (unsigned int)(K - k0), BNT, KT, (unsigned int)N);
    }
#pragma unroll
    for (int kk = 0; kk < KT; kk += 4) {
      acc0 = wmma4(ld_a(As[cur] + tr0 * 16 * KT + kk, KT, lane),
                   ld_b(Bs[cur] + kk * BNT + tc0 * 16, BNT, lane), acc0);
      acc1 = wmma4(ld_a(As[cur] + tr1 * 16 * KT + kk, KT, lane),
                   ld_b(Bs[cur] + kk * BNT + tc1 * 16, BNT, lane), acc1);
    }
    __syncthreads();  // all waves done reading cur before it is refilled
  }

  // ---- store with bias ----
  {
    int n = lane & 15, mb = (lane >> 4) << 3;
    int gc = bn + tc0 * 16 + n;
    if (gc < N) {
      float bv = bias ? bias[gc] : 0.f;
#pragma unroll
      for (int r = 0; r < 8; ++r) {
        int gr = bm + tr0 * 16 + mb + r;
        if (gr < M) C[(size_t)gr * N + gc] = acc0[r] + bv;
      }
    }
    gc = bn + tc1 * 16 + n;
    if (gc < N) {
      float bv = bias ? bias[gc] : 0.f;
#pragma unroll
      for (int r = 0; r < 8; ++r) {
        int gr = bm + tr1 * 16 + mb + r;
        if (gr < M) C[(size_t)gr * N + gc] = acc1[r] + bv;
      }
    }
  }
}

// ---------------------------------------------------------------------------
// Kernel 3: batch-norm statistics over rows -> per-column scale/shift
// ---------------------------------------------------------------------------
__global__ __launch_bounds__(256) void bn_stats_kernel(
    const float* __restrict__ Y, int Mrows, int C,
    const float* __restrict__ gamma, const float* __restrict__ beta,
    float* __restrict__ scale, float* __restrict__ shift) {
  __shared__ float ssum[256];
  __shared__ float ssq[256];
  const int col = blockIdx.x * 64 + (threadIdx.x & 63);
  const int slice = threadIdx.x >> 6;
  float s = 0.f, q = 0.f;
  if (col < C) {
    for (int r = slice; r < Mrows; r += 4) {
      float v = Y[(size_t)r * C + col];
      s += v;
      q += v * v;
    }
  }
  ssum[threadIdx.x] = s;
  ssq[threadIdx.x] = q;
  __syncthreads();
  if (slice == 0 && col < C) {
    float S = ssum[threadIdx.x] + ssum[threadIdx.x + 64] +
              ssum[threadIdx.x + 128] + ssum[threadIdx.x + 192];
    float Q = ssq[threadIdx.x] + ssq[threadIdx.x + 64] +
              ssq[threadIdx.x + 128] + ssq[threadIdx.x + 192];
    float mu = S / (float)Mrows;
    float var = Q / (float)Mrows - mu * mu;  // population var (ddof=0)
    float sc = gamma[col] * rsqrtf(var + EPS);
    scale[col] = sc;
    shift[col] = beta[col] - mu * sc;
  }
}

// ---------------------------------------------------------------------------
// Kernel 3b: fold BN1 into the next layer: Wf[k][n] = scale[k]*W[k][n],
// bias_out[n] = bias_in[n] + sum_k shift[k]*W[k][n].
// ---------------------------------------------------------------------------
__global__ __launch_bounds__(256) void fold_bn_kernel(
    const float* __restrict__ W, const float* __restrict__ bias_in,
    const float* __restrict__ scale, const float* __restrict__ shift,
    float* __restrict__ Wf, float* __restrict__ bias_out, int K, int N) {
  __shared__ float accs[256];
  const int n = blockIdx.x * 64 + (threadIdx.x & 63);
  const int slice = threadIdx.x >> 6;
  float s = 0.f;
  if (n < N) {
    for (int k = slice; k < K; k += 4) {
      float w = W[(size_t)k * N + n];
      Wf[(size_t)k * N + n] = w * scale[k];
      s += w * shift[k];
    }
  }
  accs[threadIdx.x] = s;
  __syncthreads();
  if (slice == 0 && n < N) {
    bias_out[n] = bias_in[n] + accs[threadIdx.x] + accs[threadIdx.x + 64] +
                  accs[threadIdx.x + 128] + accs[threadIdx.x + 192];
  }
}

// ---------------------------------------------------------------------------
// Kernel 4: head: logits = BN2(Y2) @ l3_w + b3; softmax over 2 classes.
// One wave32 per row; lane-parallel dot + shfl_xor reduction.
// ---------------------------------------------------------------------------
__global__ __launch_bounds__(256) void head_softmax_kernel(
    const float* __restrict__ Y2, const float* __restrict__ scale2,
    const float* __restrict__ shift2, const float* __restrict__ w3,
    const float* __restrict__ b3, float* __restrict__ out, int Mrows) {
  const int lane = threadIdx.x & 31;
  const int wave = threadIdx.x >> 5;
  const int row = blockIdx.x * 8 + wave;
  if (row >= Mrows) return;
  float d0 = 0.f, d1 = 0.f;
  for (int k = lane; k < 256; k += 32) {
    float z = Y2[(size_t)row * 256 + k] * scale2[k] + shift2[k];
    d0 += z * w3[k * 2 + 0];
    d1 += z * w3[k * 2 + 1];
  }
#pragma unroll
  for (int off = 16; off > 0; off >>= 1) {
    d0 += __shfl_xor(d0, off, 32);
    d1 += __shfl_xor(d1, off, 32);
  }
  if (lane == 0) {
    float l0 = d0 + b3[0], l1 = d1 + b3[1];
    float m = fmaxf(l0, l1);
    float e0 = __expf(l0 - m), e1 = __expf(l1 - m);
    float inv = 1.f / (e0 + e1);
    out[(size_t)row * 2 + 0] = e0 * inv;
    out[(size_t)row * 2 + 1] = e1 * inv;
  }
}

// ---------------------------------------------------------------------------
// Launch
// ---------------------------------------------------------------------------
extern "C" void kernel_launch(void* const* d_in, const int* in_sizes, int n_in,
                              void* d_out, int out_size, void* d_ws,
                              size_t ws_size, hipStream_t stream) {
  const float* input = (const float*)d_in[0];
  const float* gc1 = (const float*)d_in[1];
  const float* gc2 = (const float*)d_in[2];
  const float* l1w = (const float*)d_in[3];
  const float* l1b = (const float*)d_in[4];
  const float* bn1g = (const float*)d_in[5];
  const float* bn1b = (const float*)d_in[6];
  const float* l2w = (const float*)d_in[7];
  const float* l2b = (const float*)d_in[8];
  const float* bn2g = (const float*)d_in[9];
  const float* bn2b = (const float*)d_in[10];
  const float* l3w = (const float*)d_in[11];
  const float* l3b = (const float*)d_in[12];
  float* out = (float*)d_out;

  // workspace layout
  float* flat = (float*)d_ws;                 // [4096,1800]
  float* Y1 = flat + (size_t)B_SAMP * 1800;   // [4096,512]
  float* Y2 = Y1 + (size_t)B_SAMP * 512;      // [4096,256]
  float* sc1 = Y2 + (size_t)B_SAMP * 256;     // [512]
  float* sh1 = sc1 + 512;                     // [512]
  float* sc2 = sh1 + 512;                     // [256]
  float* sh2 = sc2 + 256;                     // [256]
  float* W2f = sh2 + 256;                     // [512,256] BN1-folded l2_w
  float* b2f = W2f + (size_t)512 * 256;       // [256]

  dim3 blk(256);

  // Stage 1: correlation + GCN, one block per sample, adj stays in LDS
  gcn_corr_kernel<<<dim3(B_SAMP), blk, 0, stream>>>(input, gc1, gc2, flat);

  // Stage 2: flat @ l1_w + b1 -> Y1 (raw; BN1 folded into next layer)
  gemm_bias_wmma<<<dim3(512 / BNT, B_SAMP / BMT), blk, 0, stream>>>(
      flat, l1w, Y1, B_SAMP, 512, 1800, l1b);
  bn_stats_kernel<<<dim3(512 / 64), blk, 0, stream>>>(Y1, B_SAMP, 512, bn1g,
                                                      bn1b, sc1, sh1);
  fold_bn_kernel<<<dim3(256 / 64), blk, 0, stream>>>(l2w, l2b, sc1, sh1, W2f,
                                                     b2f, 512, 256);

  // Stage 3: Y1 @ W2f + b2f -> Y2 (raw)
  gemm_bias_wmma<<<dim3(256 / BNT, B_SAMP / BMT), blk, 0, stream>>>(
      Y1, W2f, Y2, B_SAMP, 256, 512, b2f);
  bn_stats_kernel<<<dim3(256 / 64), blk, 0, stream>>>(Y2, B_SAMP, 256, bn2g,
                                                      bn2b, sc2, sh2);

  // Stage 4: BN2(Y2) @ l3_w + b3 -> softmax -> out
  head_softmax_kernel<<<dim3(B_SAMP / 8), blk, 0, stream>>>(Y2, sc2, sh2, l3w,
                                                            l3b, out, B_SAMP);
}